// SgcnConv_49581102465507
// MI455X (gfx1250) — compile-verified
//
#include <hip/hip_runtime.h>

// CDNA5 / gfx1250 SpMM: out[dst[e]] += val[e] * feat[src[e]]
// WMMA f32 16x16x4 computes the scaled messages D[m][n] = val[m]*feat[src[m]][n0+n]
// for a 16-edge x 16-column tile; scatter with non-returning global f32 atomics.

typedef __attribute__((ext_vector_type(2))) float v2f;
typedef __attribute__((ext_vector_type(8))) float v8f;

#define D_FEAT 128
#define WAVES_PER_BLOCK 8
#define EDGES_PER_WAVE 16
#define EDGES_PER_BLOCK (WAVES_PER_BLOCK * EDGES_PER_WAVE)

__global__ __launch_bounds__(256) void spmm_wmma_kernel(
    const float* __restrict__ feat,
    const int*   __restrict__ edge_src,
    const int*   __restrict__ edge_dst,
    const float* __restrict__ edge_val,
    float* __restrict__ out,
    int n_edges)
{
    __shared__ int   s_src[WAVES_PER_BLOCK][EDGES_PER_WAVE];
    __shared__ int   s_dst[WAVES_PER_BLOCK][EDGES_PER_WAVE];
    __shared__ float s_val[WAVES_PER_BLOCK][EDGES_PER_WAVE];

    const int lane = threadIdx.x & 31;
    const int wv   = threadIdx.x >> 5;
    const int m    = lane & 15;   // row/column index within 16-wide tile
    const int half = lane >> 4;   // 0: lanes 0-15, 1: lanes 16-31
    const int edgeBase = (blockIdx.x * WAVES_PER_BLOCK + wv) * EDGES_PER_WAVE;

    // Stage this wave's 16 edges into LDS (lanes 0-15 only).
    if (half == 0) {
        const int e = edgeBase + m;
        int src = 0, dst = 0;
        float val = 0.0f;
        if (e < n_edges) {
            src = edge_src[e];
            dst = edge_dst[e];
            val = edge_val[e];
        }
        s_src[wv][m] = src;
        s_dst[wv][m] = dst;
        s_val[wv][m] = val;
    }
    __syncthreads();

    // A-matrix layout for V_WMMA_F32_16X16X4_F32 (16x4, 2 VGPRs/lane):
    //   lanes 0-15 hold K=0 (a.x) and K=1 (a.y) for M=lane
    //   lanes 16-31 hold K=2 (a.x) and K=3 (a.y) for M=lane-16
    // For accumulating WMMA t (edges 4t..4t+3), A[m][k] = val[4t+k] * (m == 4t+k).
    const float myVal = s_val[wv][m];
    const int k0  = half * 2;      // K slot held in a.x
    const int k1  = k0 + 1;        // K slot held in a.y
    const int myT = m >> 2;        // which WMMA t this lane's edge belongs to
    const int myK = m & 3;         // this edge's K within its group of 4
    const float a0v = (myK == k0) ? myVal : 0.0f;
    const float a1v = (myK == k1) ? myVal : 0.0f;

    // B-matrix layout (4x16, 2 VGPRs/lane), mirroring A:
    //   lanes 0-15:  b.x = B[0][n=lane],    b.y = B[1][n=lane]
    //   lanes 16-31: b.x = B[2][n=lane-16], b.y = B[3][n=lane-16]
    // For WMMA t this lane gathers feat rows src[4t+k0] and src[4t+k1] at column n0+m.
    const float* rowPtr0[4];
    const float* rowPtr1[4];
#pragma unroll
    for (int t = 0; t < 4; ++t) {
        rowPtr0[t] = feat + (size_t)s_src[wv][4 * t + k0] * D_FEAT + m;
        rowPtr1[t] = feat + (size_t)s_src[wv][4 * t + k1] * D_FEAT + m;
    }

    // C/D layout: lane (half,m), VGPR v  ->  D[M = half*8 + v][N = m].
    int dstRow[8];
#pragma unroll
    for (int v = 0; v < 8; ++v) dstRow[v] = s_dst[wv][half * 8 + v];

    // Sweep all 8 column tiles, reusing the staged edge metadata.
#pragma unroll 1
    for (int n0 = 0; n0 < D_FEAT; n0 += 16) {
        v8f c = {};
#pragma unroll
        for (int t = 0; t < 4; ++t) {
            v2f a;
            a.x = (myT == t) ? a0v : 0.0f;
            a.y = (myT == t) ? a1v : 0.0f;
            v2f b;
            b.x = rowPtr0[t][n0];   // L2-resident gather (feat fits in 192MB L2)
            b.y = rowPtr1[t][n0];
            // D = A*B + C, fp32 exact: D[m][n] accumulates val[m]*feat[src[m]][n0+n]
            c = __builtin_amdgcn_wmma_f32_16x16x4_f32(
                    false, a, false, b, (short)0, c, false, false);
        }
        const int n = n0 + m;
#pragma unroll
        for (int v = 0; v < 8; ++v) {
            // non-returning global_atomic_add_f32 (fire-and-forget, STOREcnt)
            atomicAdd(out + (size_t)dstRow[v] * D_FEAT + n, c[v]);
        }
    }
}

extern "C" void kernel_launch(void* const* d_in, const int* in_sizes, int n_in,
                              void* d_out, int out_size, void* d_ws, size_t ws_size,
                              hipStream_t stream) {
    const float* feat     = (const float*)d_in[0];
    const int*   edge_src = (const int*)d_in[1];
    const int*   edge_dst = (const int*)d_in[2];
    const float* edge_val = (const float*)d_in[3];
    float* out = (float*)d_out;
    const int n_edges = in_sizes[1];

    // Atomic-accumulate kernel: output must start at zero every call.
    hipMemsetAsync(d_out, 0, (size_t)out_size * sizeof(float), stream);

    const int blocks = (n_edges + EDGES_PER_BLOCK - 1) / EDGES_PER_BLOCK;
    spmm_wmma_kernel<<<blocks, 256, 0, stream>>>(feat, edge_src, edge_dst,
                                                 edge_val, out, n_edges);
}